// MultiDense_20701742367019
// MI455X (gfx1250) — compile-verified
//
#include <hip/hip_runtime.h>

// ---------------- problem constants ----------------
#define DIMS      64
#define IN_DIM    128
#define OUT_DIM   128
#define BATCH     16384
#define SEQ       4

// ---------------- tiling ----------------
#define SAMPLES_PER_TILE  16          // 16 samples * 4 rows = 64 M-rows = 4 waves * M16
#define WAVES_PER_BLOCK   4
#define THREADS_PER_BLOCK 128
#define MAX_TILES         (BATCH / SAMPLES_PER_TILE + DIMS)   // 1088 upper bound

// LDS layout (in dwords).
//  W  : row-major [K=128][N=128], row stride 132 dwords (TDM pads 4 dwords / 128)
//  A  : per-wave [row=16][K=128], row stride 132 dwords (conflict-free b64 reads)
#define WT_STRIDE   132
#define WT_DWORDS   (IN_DIM * WT_STRIDE)             // 128*132 = 16896
#define A_STRIDE    132
#define A_DWORDS    (16 * A_STRIDE)
#define LDS_DWORDS  (WT_DWORDS + WAVES_PER_BLOCK * A_DWORDS)
#define LDS_BYTES   (LDS_DWORDS * 4)                 // 101376 bytes

typedef float v2f __attribute__((ext_vector_type(2)));
typedef float v8f __attribute__((ext_vector_type(8)));
typedef unsigned int v4u __attribute__((ext_vector_type(4)));
typedef int v4i_t __attribute__((ext_vector_type(4)));
typedef int v8i_t __attribute__((ext_vector_type(8)));

// ---------------- workspace layout (ints) ----------------
// [0,64) counts | [64,129) seg_start | [129,194) tile_base | [194,258) cursor | [258,..) order[B]

__global__ void md_zero(int* __restrict__ counts, int* __restrict__ cursor) {
    const int i = threadIdx.x;
    if (i < DIMS) { counts[i] = 0; cursor[i] = 0; }
}

__global__ void md_hist(const int* __restrict__ lookups, int* __restrict__ counts) {
    const int i = blockIdx.x * blockDim.x + threadIdx.x;
    if (i < BATCH) atomicAdd(&counts[lookups[i] & (DIMS - 1)], 1);
}

__global__ void md_scan(const int* __restrict__ counts,
                        int* __restrict__ seg_start, int* __restrict__ tile_base) {
    if (threadIdx.x == 0) {
        int s = 0, t = 0;
        for (int e = 0; e < DIMS; ++e) {
            seg_start[e] = s; tile_base[e] = t;
            s += counts[e];
            t += (counts[e] + SAMPLES_PER_TILE - 1) / SAMPLES_PER_TILE;
        }
        seg_start[DIMS] = s; tile_base[DIMS] = t;
    }
}

__global__ void md_scatter(const int* __restrict__ lookups,
                           const int* __restrict__ seg_start,
                           int* __restrict__ cursor, int* __restrict__ order) {
    const int i = blockIdx.x * blockDim.x + threadIdx.x;
    if (i < BATCH) {
        const int e = lookups[i] & (DIMS - 1);
        const int pos = atomicAdd(&cursor[e], 1);
        order[seg_start[e] + pos] = i;
    }
}

__global__ __launch_bounds__(THREADS_PER_BLOCK)
void md_gemm(const float* __restrict__ values, const float* __restrict__ W,
             const float* __restrict__ bias,
             const int* __restrict__ seg_start, const int* __restrict__ tile_base,
             const int* __restrict__ order, float* __restrict__ out)
{
    extern __shared__ float lds[];
    const int tid  = threadIdx.x;
    const int wave = tid >> 5;
    const int lane = tid & 31;

    const int t = blockIdx.x;
    if (t >= tile_base[DIMS]) return;           // uniform exit, EXEC stays full

    // locate expert segment for this tile (tile_base is non-decreasing)
    int e = 0;
    for (int i = 1; i < DIMS; ++i) e += (t >= tile_base[i]) ? 1 : 0;

    const int seg0  = seg_start[e];
    const int cnt   = seg_start[e + 1] - seg0;
    const int slot0 = (t - tile_base[e]) * SAMPLES_PER_TILE;

    float* Wt    = lds;                                     // [K][N], stride 132
    float* Arows = lds + WT_DWORDS + wave * A_DWORDS;       // 16 rows x 128 K

    // ---- fire TDM: DMA W[e] (128x128 f32) L2 -> LDS with 4-dword/row padding ----
    if (wave == 0) {
        const unsigned lds_base = (unsigned)(unsigned long long)(const void*)Wt;
        const unsigned long long ga =
            (unsigned long long)(const void*)(W + (size_t)e * IN_DIM * OUT_DIM);
        // D# group0: count=1 | lds_addr | global_addr[56:0] | type=2
        const v4u g0 = { 1u,
                         lds_base,
                         (unsigned)(ga & 0xFFFFFFFFu),
                         (unsigned)(((ga >> 32) & 0x01FFFFFFu) | (2u << 30)) };
        // D# group1: data_size=2(4B), pad_enable, pad_interval=6 (128 dw),
        //            pad_amount=3 (4 dw); tensor 128x128; tile 128x128; stride0=128
        const v8i_t g1 = { (int)((2u << 16) | (1u << 20) | (6u << 22) | (3u << 25)),
                           (int)(128u << 16),          // tensor_dim0 = 128
                           (int)(128u << 16),          // tensor_dim1 = 128
                           (int)(128u << 16),          // tile_dim0   = 128
                           128,                        // tile_dim1   = 128, tile_dim2 = 0
                           128,                        // tensor_dim0_stride = 128
                           (int)(16384u << 16),        // tensor_dim1_stride = 16384
                           0 };
        const v4i_t g2 = { 0, 0, 0, 0 };
        const v4i_t g3 = { 0, 0, 0, 0 };
        const v8i_t g4 = { 0, 0, 0, 0, 0, 0, 0, 0 };   // 6-arg toolchain: extra group
        __builtin_amdgcn_tensor_load_to_lds(g0, g1, g2, g3, g4, 0);
    }

    // ---- stage this wave's A tile: 4 samples x SEQ rows x 128 (overlaps TDM) ----
    int samp_of[4];
    #pragma unroll
    for (int j = 0; j < 4; ++j) {
        const int slot = slot0 + wave * 4 + j;
        samp_of[j] = order[seg0 + (slot < cnt ? slot : 0)];  // clamp invalid -> sample 0
    }
    #pragma unroll
    for (int j = 0; j < 16; ++j) {
        const float* src = values + ((size_t)samp_of[j >> 2] * SEQ + (j & 3)) * IN_DIM;
        const float4 v = ((const float4*)src)[lane];         // b128, coalesced
        *(float4*)(Arows + j * A_STRIDE + lane * 4) = v;     // 16B-aligned ds_store_b128
    }

    if (wave == 0) __builtin_amdgcn_s_wait_tensorcnt(0);     // TDM data landed
    __syncthreads();

    // ---- WMMA f32 16x16x4 main loop ----
    // A frag: lanes 0-15 row M=lane, K=4k,4k+1 ; lanes 16-31 K=4k+2,4k+3
    // B frag: lanes 0-15 N=lane,  rows K'=4k (comp0) / 4k+1 (comp1); upper half K'+2
    const int c2  = (lane >> 4) * 2;
    const int row = lane & 15;
    const int nl  = lane & 15;
    const float* Ab = Arows + row * A_STRIDE;
    const float* Bb = Wt + nl;

    v8f acc[8] = {};
    for (int k = 0; k < IN_DIM; k += 4) {
        const v2f a = *(const v2f*)(Ab + k + c2);
        const float* brow = Bb + (k + c2) * WT_STRIDE;
        #pragma unroll
        for (int n = 0; n < 8; ++n) {
            const v2f bf = { brow[n * 16], brow[n * 16 + WT_STRIDE] };
            acc[n] = __builtin_amdgcn_wmma_f32_16x16x4_f32(
                /*neg_a=*/false, a, /*neg_b=*/false, bf,
                /*c_mod=*/(short)0, acc[n], /*reuse_a=*/false, /*reuse_b=*/false);
        }
    }

    // ---- bias + scatter store ----
    const int half = lane >> 4;
    int  samp_r[8];
    bool valid_r[8];
    #pragma unroll
    for (int r = 0; r < 8; ++r) {
        const int m    = r + half * 8;
        const int slot = slot0 + wave * 4 + (m >> 2);
        valid_r[r] = (slot < cnt);
        samp_r[r]  = samp_of[m >> 2];
    }
    for (int n = 0; n < 8; ++n) {
        const float bv = bias[e * OUT_DIM + n * 16 + nl];
        #pragma unroll
        for (int r = 0; r < 8; ++r) {
            if (valid_r[r]) {
                const int m = r + half * 8;
                out[((size_t)samp_r[r] * SEQ + (m & 3)) * OUT_DIM + n * 16 + nl]
                    = acc[n][r] + bv;
            }
        }
    }
}

extern "C" void kernel_launch(void* const* d_in, const int* in_sizes, int n_in,
                              void* d_out, int out_size, void* d_ws, size_t ws_size,
                              hipStream_t stream) {
    (void)in_sizes; (void)n_in; (void)out_size; (void)ws_size;
    const float* values  = (const float*)d_in[0];
    const float* W       = (const float*)d_in[1];
    const float* bias    = (const float*)d_in[2];
    const int*   lookups = (const int*)d_in[3];
    float* out = (float*)d_out;

    int* ws        = (int*)d_ws;
    int* counts    = ws;
    int* seg_start = ws + 64;
    int* tile_base = ws + 129;
    int* cursor    = ws + 194;
    int* order     = ws + 258;

    (void)hipFuncSetAttribute((const void*)md_gemm,
                              hipFuncAttributeMaxDynamicSharedMemorySize, LDS_BYTES);

    md_zero   <<<1, 256, 0, stream>>>(counts, cursor);
    md_hist   <<<BATCH / 256, 256, 0, stream>>>(lookups, counts);
    md_scan   <<<1, 32, 0, stream>>>(counts, seg_start, tile_base);
    md_scatter<<<BATCH / 256, 256, 0, stream>>>(lookups, seg_start, cursor, order);
    md_gemm   <<<MAX_TILES, THREADS_PER_BLOCK, LDS_BYTES, stream>>>(
        values, W, bias, seg_start, tile_base, order, out);
}